// MultiscalePadicLoss_12335146074798
// MI455X (gfx1250) — compile-verified
//
#include <hip/hip_runtime.h>
#include <hip/hip_bf16.h>

typedef float v2f __attribute__((ext_vector_type(2)));
typedef float v8f __attribute__((ext_vector_type(8)));

#define NPTS 4096
#define DIM  512
#define TILE 128
#define KC   32
#define LROW 36   // padded LDS row stride (floats); 144B row pitch keeps b64 reads conflict-free

#if __has_builtin(__builtin_amdgcn_global_load_async_to_lds_b128)
#define HAVE_ASYNC_LDS 1
typedef int v4i_vs __attribute__((vector_size(16)));
typedef __attribute__((address_space(1))) v4i_vs* gas_ptr_t;   // global (AS1) 16B-vector ptr
typedef __attribute__((address_space(3))) v4i_vs* las_ptr_t;   // LDS    (AS3) 16B-vector ptr
#else
#define HAVE_ASYNC_LDS 0
#endif

__device__ __forceinline__ void wait_async_zero() {
    asm volatile("s_wait_asynccnt 0x0" ::: "memory");
}

// ---------------- kernel 1: zero global accumulators ----------------
__global__ void padic_init_kernel(float* gsum, unsigned* gcnt) {
    int t = threadIdx.x;
    if (t < 8) { gsum[t] = 0.0f; gcnt[t] = 0u; }
}

// ---------------- kernel 2: row squared norms (one wave32 per row) ----------------
__global__ __launch_bounds__(256) void padic_rownorm_kernel(const float* __restrict__ z,
                                                            float* __restrict__ sq) {
    int row  = blockIdx.x * 8 + (threadIdx.x >> 5);
    int lane = threadIdx.x & 31;
    const float4* zr = (const float4*)(z + (size_t)row * DIM);
    float acc = 0.0f;
    #pragma unroll
    for (int c = 0; c < (DIM / 4) / 32; ++c) {
        float4 v = zr[lane + c * 32];
        acc += v.x * v.x + v.y * v.y + v.z * v.z + v.w * v.w;
    }
    #pragma unroll
    for (int off = 16; off > 0; off >>= 1) acc += __shfl_down(acc, off, 32);
    if (lane == 0) sq[row] = acc;
}

// ---------------- kernel 3: fused Gram tile + p-adic level accumulation ----------------
__global__ __launch_bounds__(256) void padic_gram_kernel(
    const float* __restrict__ z, const int* __restrict__ idx,
    const float* __restrict__ sq, float* __restrict__ gsum, unsigned* __restrict__ gcnt)
{
    const int tj = blockIdx.x;   // column tile
    const int ti = blockIdx.y;   // row tile
    if (tj < ti) return;         // upper-triangular tiles only (uniform per block)

    __shared__ __align__(16) float As[2][TILE][LROW];
    __shared__ __align__(16) float Bs[2][TILE][LROW];
    __shared__ int      idxR[TILE], idxC[TILE];
    __shared__ float    sqR[TILE],  sqC[TILE];
    __shared__ float    sSum[6];
    __shared__ unsigned sCnt[6];

    const int t = threadIdx.x;
    const int iBase = ti * TILE, jBase = tj * TILE;
    if (t < 6) { sSum[t] = 0.0f; sCnt[t] = 0u; }
    if (t < TILE) {
        idxR[t] = idx[iBase + t];
        sqR[t]  = sq[iBase + t];
        idxC[t] = idx[jBase + t];
        sqC[t]  = sq[jBase + t];
    }

    const int wave  = t >> 5, lane = t & 31;
    const int waveM = (wave >> 1) * 32;    // 0,32,64,96
    const int waveN = (wave & 1) * 64;     // 0,64
    const int lm    = lane & 15;           // M/N selector within 16-wide subtile
    const int khalf = (lane >> 4) * 2;     // K pair: lanes 0-15 -> k{0,1}, 16-31 -> k{2,3}

    v8f acc[2][4] = {};

#if HAVE_ASYNC_LDS
    // ---- async global->LDS staging of one K-chunk into buffer `buf` ----
    auto stage = [&](int buf, int kc) {
        #pragma unroll
        for (int it = 0; it < 4; ++it) {
            int id = t + it * 256;
            int r  = id >> 3;
            int c4 = (id & 7) * 4;
            v4i_vs* pa = (v4i_vs*)(z + (size_t)(iBase + r) * DIM + kc + c4);
            v4i_vs* pb = (v4i_vs*)(z + (size_t)(jBase + r) * DIM + kc + c4);
            __builtin_amdgcn_global_load_async_to_lds_b128((gas_ptr_t)pa, (las_ptr_t)(v4i_vs*)&As[buf][r][c4], 0, 0);
            __builtin_amdgcn_global_load_async_to_lds_b128((gas_ptr_t)pb, (las_ptr_t)(v4i_vs*)&Bs[buf][r][c4], 0, 0);
        }
    };
    stage(0, 0);
    wait_async_zero();
    __syncthreads();
#endif

    for (int kc = 0; kc < DIM; kc += KC) {
        const int cur = (kc / KC) & 1;
#if HAVE_ASYNC_LDS
        if (kc + KC < DIM) stage(cur ^ 1, kc + KC);   // overlap next-chunk DMA with WMMA
#else
        if (kc) __syncthreads();                      // LDS reuse guard
        #pragma unroll
        for (int it = 0; it < 4; ++it) {
            int id = t + it * 256;
            int r  = id >> 3;
            int c4 = (id & 7) * 4;
            const float* pa = z + (size_t)(iBase + r) * DIM + kc + c4;
            const float* pb = z + (size_t)(jBase + r) * DIM + kc + c4;
            *(float4*)&As[cur][r][c4] = *(const float4*)pa;
            *(float4*)&Bs[cur][r][c4] = *(const float4*)pb;
            if (kc + KC < DIM) {
                __builtin_prefetch(pa + KC, 0, 1);
                __builtin_prefetch(pb + KC, 0, 1);
            }
        }
        __syncthreads();
#endif

        #pragma unroll
        for (int ks = 0; ks < KC; ks += 4) {
            const int kk = ks + khalf;
            v2f a0 = *(const v2f*)&As[cur][waveM      + lm][kk];
            v2f a1 = *(const v2f*)&As[cur][waveM + 16 + lm][kk];
            v2f b0 = *(const v2f*)&Bs[cur][waveN      + lm][kk];
            v2f b1 = *(const v2f*)&Bs[cur][waveN + 16 + lm][kk];
            v2f b2 = *(const v2f*)&Bs[cur][waveN + 32 + lm][kk];
            v2f b3 = *(const v2f*)&Bs[cur][waveN + 48 + lm][kk];
            acc[0][0] = __builtin_amdgcn_wmma_f32_16x16x4_f32(false, a0, false, b0, (short)0, acc[0][0], false, false);
            acc[0][1] = __builtin_amdgcn_wmma_f32_16x16x4_f32(false, a0, false, b1, (short)0, acc[0][1], false, false);
            acc[0][2] = __builtin_amdgcn_wmma_f32_16x16x4_f32(false, a0, false, b2, (short)0, acc[0][2], false, false);
            acc[0][3] = __builtin_amdgcn_wmma_f32_16x16x4_f32(false, a0, false, b3, (short)0, acc[0][3], false, false);
            acc[1][0] = __builtin_amdgcn_wmma_f32_16x16x4_f32(false, a1, false, b0, (short)0, acc[1][0], false, false);
            acc[1][1] = __builtin_amdgcn_wmma_f32_16x16x4_f32(false, a1, false, b1, (short)0, acc[1][1], false, false);
            acc[1][2] = __builtin_amdgcn_wmma_f32_16x16x4_f32(false, a1, false, b2, (short)0, acc[1][2], false, false);
            acc[1][3] = __builtin_amdgcn_wmma_f32_16x16x4_f32(false, a1, false, b3, (short)0, acc[1][3], false, false);
        }

#if HAVE_ASYNC_LDS
        if (kc + KC < DIM) {
            wait_async_zero();   // next buffer fully resident in LDS
            __syncthreads();     // ...and all waves done reading current buffer
        }
#endif
    }
    __syncthreads();

    // ----- fused epilogue: distances + 3-adic levels, register accumulation -----
    float lsum[6] = {0, 0, 0, 0, 0, 0};
    int   lcnt[6] = {0, 0, 0, 0, 0, 0};
    const float TARG[6] = {1.0f, 1.0f / 3.0f, 1.0f / 9.0f, 1.0f / 27.0f, 1.0f / 81.0f, 1.0f / 243.0f};
    const int rowHalf = (lane >> 4) * 8;   // C layout: lanes 16-31 hold M = r+8

    #pragma unroll
    for (int s = 0; s < 2; ++s) {
        #pragma unroll
        for (int u = 0; u < 4; ++u) {
            #pragma unroll
            for (int r = 0; r < 8; ++r) {
                int mi = waveM + s * 16 + r + rowHalf;
                int nj = waveN + u * 16 + lm;
                int gi = iBase + mi, gj = jBase + nj;
                if (gj > gi) {
                    int dv = idxR[mi] - idxC[nj];
                    int diff = dv < 0 ? -dv : dv;
                    if (diff > 0) {
                        int val = (diff % 3 == 0) + (diff % 9 == 0) + (diff % 27 == 0)
                                + (diff % 81 == 0) + (diff % 243 == 0) + (diff % 729 == 0);
                        if (val < 6) {
                            float d2 = sqR[mi] + sqC[nj] - 2.0f * acc[s][u][r];
                            float dist = d2 > 0.0f ? sqrtf(d2) : 0.0f;
                            #pragma unroll
                            for (int v = 0; v < 6; ++v) {
                                if (val == v) {
                                    float e = dist - TARG[v];
                                    lsum[v] += e * e;
                                    lcnt[v] += 1;
                                }
                            }
                        }
                    }
                }
            }
        }
    }

    #pragma unroll
    for (int v = 0; v < 6; ++v) {
        if (lcnt[v] > 0) {
            atomicAdd(&sSum[v], lsum[v]);
            atomicAdd(&sCnt[v], (unsigned)lcnt[v]);
        }
    }
    __syncthreads();
    if (t < 6 && sCnt[t] > 0u) {
        atomicAdd(&gsum[t], sSum[t]);
        atomicAdd(&gcnt[t], sCnt[t]);
    }
}

// ---------------- kernel 4: finalize scalar ----------------
__global__ void padic_finalize_kernel(const float* __restrict__ gsum,
                                      const unsigned* __restrict__ gcnt,
                                      float* __restrict__ out) {
    if (threadIdx.x == 0) {
        float total = 0.0f, levels = 0.0f;
        #pragma unroll
        for (int v = 0; v < 6; ++v) {
            unsigned c = gcnt[v];
            float mse = gsum[v] / fmaxf((float)c, 1.0f);
            if (c >= 2u) {
                total += mse / (float)(v + 1);
                levels += 1.0f;
            }
        }
        out[0] = total / fmaxf(levels, 1.0f);
    }
}

extern "C" void kernel_launch(void* const* d_in, const int* in_sizes, int n_in,
                              void* d_out, int out_size, void* d_ws, size_t ws_size,
                              hipStream_t stream) {
    const float* z   = (const float*)d_in[0];
    const int*   idx = (const int*)d_in[1];

    float*    sq   = (float*)d_ws;                                   // NPTS floats
    float*    gsum = (float*)((char*)d_ws + (size_t)NPTS * sizeof(float));
    unsigned* gcnt = (unsigned*)(gsum + 8);

    padic_init_kernel<<<1, 64, 0, stream>>>(gsum, gcnt);
    padic_rownorm_kernel<<<NPTS / 8, 256, 0, stream>>>(z, sq);
    dim3 grid(NPTS / TILE, NPTS / TILE);
    padic_gram_kernel<<<grid, 256, 0, stream>>>(z, idx, sq, gsum, gcnt);
    padic_finalize_kernel<<<1, 32, 0, stream>>>(gsum, gcnt, (float*)d_out);
}